// HyperbolicMessagePassing_2508260901394
// MI455X (gfx1250) — compile-verified
//
#include <hip/hip_runtime.h>
#include <math.h>

#define NN 10000
#define NE 160000
#define DD 128
#define EPSF 1e-7f

typedef __attribute__((ext_vector_type(2))) float v2f;
typedef __attribute__((ext_vector_type(8))) float v8f;

static __device__ __forceinline__ float wred(float v) {
  #pragma unroll
  for (int m = 16; m >= 1; m >>= 1) v += __shfl_xor(v, m, 32);
  return v;
}

__global__ void k_zero_i32(int* __restrict__ p, int n) {
  int i = blockIdx.x * blockDim.x + threadIdx.x;
  if (i < n) p[i] = 0;
}

// Per-node squared norms s2[i] = ||x_i||^2
__global__ void k_norms(const float* __restrict__ x, float* __restrict__ norms) {
  int i = blockIdx.x * blockDim.x + threadIdx.x;
  if (i >= NN) return;
  const float4* row = (const float4*)(x + (size_t)i * DD);
  float s = 0.f;
  #pragma unroll 8
  for (int j = 0; j < DD / 4; j++) {
    float4 v = row[j];
    s += v.x * v.x + v.y * v.y + v.z * v.z + v.w * v.w;
  }
  norms[i] = s;
}

// Per-edge dots sd[e] = x[src[e]] . x[dst[e]] via chained V_WMMA_F32_16X16X4_F32.
// One wave handles 16 edges: A = 16 src rows (16xK), B = 16 dst rows as columns
// (Kx16); diagonal of the 16x16 accumulator gives the 16 dots.
__global__ void k_edge_dots(const float* __restrict__ x, const int* __restrict__ src,
                            const int* __restrict__ dst, float* __restrict__ sd) {
  int wave = (blockIdx.x * blockDim.x + threadIdx.x) >> 5;
  int lane = threadIdx.x & 31;
  int base = wave * 16;
  if (base >= NE) return;  // wave-uniform: EXEC stays all-ones for WMMA
  int e = base + (lane & 15);             // lane&15 = row M of A and col N of B
  const float* arow = x + (size_t)src[e] * DD;
  const float* brow = x + (size_t)dst[e] * DD;
  int co = (lane >> 4) * 2;               // lanes 0-15 hold K={k,k+1}, 16-31 K={k+2,k+3}
  v8f c = {};
  for (int k = 0; k < DD; k += 4) {
    v2f a = *(const v2f*)(arow + k + co);
    v2f b = *(const v2f*)(brow + k + co);
    c = __builtin_amdgcn_wmma_f32_16x16x4_f32(false, a, false, b, (short)0, c,
                                              false, false);
  }
  float cc[8];
  #pragma unroll
  for (int j = 0; j < 8; j++) cc[j] = c[j];
  // C/D layout: VGPR r, lane L -> M = r + 8*(L/16), N = L%16. Diagonal (i,i):
  // i<8  -> lane i,    component i
  // i>=8 -> lane i+16, component i-8
  if (lane < 8)       sd[base + lane] = cc[lane];
  else if (lane >= 24) sd[base + (lane - 16)] = cc[lane - 24];
}

__global__ void k_count(const int* __restrict__ dst, int* __restrict__ counts) {
  int e = blockIdx.x * blockDim.x + threadIdx.x;
  if (e < NE) atomicAdd(&counts[dst[e]], 1);
}

// Exclusive prefix sum over NN counts, single wave32.
__global__ void k_scan(const int* __restrict__ counts, int* __restrict__ starts) {
  int lane = threadIdx.x;
  int running = 0;
  for (int b = 0; b < NN; b += 32) {
    int i = b + lane;
    int cv = (i < NN) ? counts[i] : 0;
    int v = cv;
    #pragma unroll
    for (int off = 1; off < 32; off <<= 1) {
      int n = __shfl_up(v, off, 32);
      if (lane >= off) v += n;
    }
    if (i < NN) starts[i] = running + v - cv;
    running += __shfl(v, 31, 32);
  }
  if (lane == 0) starts[NN] = running;
}

__global__ void k_fill(const int* __restrict__ dst, const int* __restrict__ starts,
                       int* __restrict__ cursor, int* __restrict__ elist) {
  int e = blockIdx.x * blockDim.x + threadIdx.x;
  if (e >= NE) return;
  int d = dst[e];
  int p = atomicAdd(&cursor[d], 1);
  elist[starts[d] + p] = e;
}

// Restore ascending edge-id order per node (the fold is non-associative).
__global__ void k_sort(const int* __restrict__ starts, int* __restrict__ elist) {
  int i = blockIdx.x * blockDim.x + threadIdx.x;
  if (i >= NN) return;
  int s0 = starts[i], s1 = starts[i + 1];
  for (int a = s0 + 1; a < s1; a++) {
    int v = elist[a];
    int b = a - 1;
    while (b >= s0 && elist[b] > v) { elist[b + 1] = elist[b]; b--; }
    elist[b + 1] = v;
  }
}

// One wave32 per node: sequential Mobius fold over its incident edges,
// then Mobius scalar-mean and final mobius_add(x, aggr).
__global__ void __launch_bounds__(256)
k_fold(const float* __restrict__ x, const int* __restrict__ src,
       const float* __restrict__ norms, const float* __restrict__ sd,
       const int* __restrict__ starts, const int* __restrict__ elist,
       float* __restrict__ out) {
  int node = (blockIdx.x * blockDim.x + threadIdx.x) >> 5;
  int lane = threadIdx.x & 31;
  if (node >= NN) return;

  const float* xrow = x + (size_t)node * DD;
  float xd[4], acc[4];
  #pragma unroll
  for (int j = 0; j < 4; j++) { xd[j] = xrow[lane + 32 * j]; acc[j] = 0.f; }
  float d2 = norms[node];
  float a2 = 0.f;

  int s0 = starts[node], s1 = starts[node + 1];
  for (int t = s0; t < s1; ++t) {
    int e = elist[t];
    int sn = src[e];
    const float* srow = x + (size_t)sn * DD;
    float s2 = norms[sn];
    float dsd = sd[e];
    // message m = mobius_add(xs, -xd) = alpha*xs + beta*xd (scalars only)
    float den_m = fmaxf(1.f - 2.f * dsd + s2 * d2, EPSF);
    float alpha = (1.f - 2.f * dsd + d2) / den_m;
    float beta = -(1.f - s2) / den_m;
    float m[4];
    float am_p = 0.f, m2_p = 0.f;
    #pragma unroll
    for (int j = 0; j < 4; j++) {
      float xs = srow[lane + 32 * j];
      m[j] = alpha * xs + beta * xd[j];
      am_p += acc[j] * m[j];
      m2_p += m[j] * m[j];
    }
    float am = wred(am_p);
    float m2 = wred(m2_p);
    // acc = mobius_add(acc, m)
    float A = 1.f + 2.f * am + m2;
    float B = 1.f - a2;
    float den = fmaxf(1.f + 2.f * am + a2 * m2, EPSF);
    float inv = 1.f / den;
    float a2_p = 0.f;
    #pragma unroll
    for (int j = 0; j < 4; j++) {
      acc[j] = (A * acc[j] + B * m[j]) * inv;
      a2_p += acc[j] * acc[j];
    }
    a2 = wred(a2_p);
  }

  // mobius_scalar_mul(1/max(count,1), acc)
  float cnt = (float)(s1 - s0);
  float r = 1.f / fmaxf(cnt, 1.f);
  float nrm = sqrtf(fmaxf(a2, 0.f));
  float nrm_c = fminf(fmaxf(nrm, EPSF), 1.f - EPSF);
  float g = tanhf(r * atanhf(nrm_c)) / nrm_c;
  if (!(nrm > EPSF)) g = 0.f;
  float aggr[4];
  float xy_p = 0.f;
  #pragma unroll
  for (int j = 0; j < 4; j++) {
    aggr[j] = g * acc[j];
    xy_p += xd[j] * aggr[j];
  }
  float xy = wred(xy_p);
  float y2 = g * g * a2;

  // out = mobius_add(x, aggr)
  float A = 1.f + 2.f * xy + y2;
  float B = 1.f - d2;
  float den = fmaxf(1.f + 2.f * xy + d2 * y2, EPSF);
  float inv = 1.f / den;
  float* orow = out + (size_t)node * DD;
  #pragma unroll
  for (int j = 0; j < 4; j++) orow[lane + 32 * j] = (A * xd[j] + B * aggr[j]) * inv;
}

extern "C" void kernel_launch(void* const* d_in, const int* in_sizes, int n_in,
                              void* d_out, int out_size, void* d_ws, size_t ws_size,
                              hipStream_t stream) {
  const float* x = (const float*)d_in[0];
  const int* ei = (const int*)d_in[1];  // edge_index [2, NE] row-major
  const int* src = ei;
  const int* dst = ei + NE;
  float* out = (float*)d_out;

  char* w = (char*)d_ws;
  size_t o = 0;
  auto wsalloc = [&](size_t bytes) -> char* {
    char* p = w + o;
    o = (o + bytes + 255) & ~(size_t)255;
    return p;
  };
  float* norms = (float*)wsalloc(NN * sizeof(float));
  float* sd    = (float*)wsalloc(NE * sizeof(float));
  int* counts  = (int*)wsalloc(NN * sizeof(int));
  int* starts  = (int*)wsalloc((NN + 1) * sizeof(int));
  int* cursor  = (int*)wsalloc(NN * sizeof(int));
  int* elist   = (int*)wsalloc(NE * sizeof(int));

  const int B = 256;
  k_zero_i32<<<(NN + B - 1) / B, B, 0, stream>>>(counts, NN);
  k_zero_i32<<<(NN + B - 1) / B, B, 0, stream>>>(cursor, NN);
  k_norms<<<(NN + B - 1) / B, B, 0, stream>>>(x, norms);
  // 16 edges per wave -> NE/16 waves, 8 waves/block
  k_edge_dots<<<((NE / 16) * 32 + B - 1) / B, B, 0, stream>>>(x, src, dst, sd);
  k_count<<<(NE + B - 1) / B, B, 0, stream>>>(dst, counts);
  k_scan<<<1, 32, 0, stream>>>(counts, starts);
  k_fill<<<(NE + B - 1) / B, B, 0, stream>>>(dst, starts, cursor, elist);
  k_sort<<<(NN + B - 1) / B, B, 0, stream>>>(starts, elist);
  k_fold<<<(NN * 32 + B - 1) / B, B, 0, stream>>>(x, src, norms, sd, starts, elist, out);
}